// ChamferDistance_65987877535942
// MI455X (gfx1250) — compile-verified
//
#include <hip/hip_runtime.h>

typedef float v2f __attribute__((ext_vector_type(2)));
typedef float v8f __attribute__((ext_vector_type(8)));

constexpr int BATCH = 4;
constexpr int NPTS  = 8192;
constexpr int ROW_TILES_PER_WAVE = 4;                        // 64 query rows / wave
constexpr int WAVES_PER_BLOCK    = 8;                        // 256 threads
constexpr int ROW_GROUPS = NPTS / (16 * ROW_TILES_PER_WAVE); // 128 per (b,dir)
constexpr int NT = NPTS / 16;                                // 512 column tiles

// Single-instruction float min (skip IEEE canonicalization: inputs are NaN-free)
__device__ __forceinline__ float fmin_asm(float a, float b) {
    float r;
    asm("v_min_num_f32 %0, %1, %2" : "=v"(r) : "v"(a), "v"(b));
    return r;
}

// ---------------------------------------------------------------------------
// Preprocess: per batch, per cloud, store pair arrays (float2, 8B stores):
//   xy[n] = (x, y)          zs[n] = (z, x^2+y^2+z^2)
// ws layout (float2 units), per batch: [c0xy | c0zs | c1xy | c1zs], N each.
// cloud0 = gt cloud (coords+gt), cloud1 = gen cloud (coords+pred).
// Total ws: 4 batches * 4N float2 = 1 MB.
// ---------------------------------------------------------------------------
__global__ void chamfer_pre(const float* __restrict__ pred,
                            const float* __restrict__ gt,
                            const float* __restrict__ coords,
                            v2f* __restrict__ ws) {
    int i = blockIdx.x * blockDim.x + threadIdx.x;
    if (i >= BATCH * NPTS) return;
    int b = i / NPTS, n = i % NPTS;
    const int base = b * 3 * NPTS + n;                 // [B, C, N] planar input
    float cx = coords[base], cy = coords[base + NPTS], cz = coords[base + 2 * NPTS];
    float x1 = cx + gt[base];
    float y1 = cy + gt[base + NPTS];
    float z1 = cz + gt[base + 2 * NPTS];
    float x2 = cx + pred[base];
    float y2 = cy + pred[base + NPTS];
    float z2 = cz + pred[base + 2 * NPTS];
    v2f* p = ws + (size_t)b * 4 * NPTS;
    p[0 * NPTS + n] = (v2f){x1, y1};
    p[1 * NPTS + n] = (v2f){z1, x1 * x1 + y1 * y1 + z1 * z1};
    p[2 * NPTS + n] = (v2f){x2, y2};
    p[3 * NPTS + n] = (v2f){z2, x2 * x2 + y2 * y2 + z2 * z2};
}

// ---------------------------------------------------------------------------
// Main: one wave = 64 query rows of one direction of one batch.
// D = A x B, A[m] = (-2x_q, -2y_q, -2z_q, 1), B[:,n] = (x_t, y_t, z_t, sq_t)
// => D[m][n] = -2*dot + |p_t|^2.  Running min over n kept in registers;
// |p_q|^2 added after the reduction.  B fragments are software-pipelined:
// the next tile's global_load_b64 is issued before the current tile's WMMAs.
//
// f32 16x16x4 fragment layout (wave32):
//   A (v2f): lanes 0-15 -> (K0,K1) of row M=lane; lanes 16-31 -> (K2,K3)
//   B (v2f): lanes 0-15 -> (K0,K1) of col N=lane; lanes 16-31 -> (K2,K3)
//   C/D (v8f): VGPR r, lanes 0-15 -> row r, lanes 16-31 -> row r+8, col = lane%16
// ---------------------------------------------------------------------------
__global__ __launch_bounds__(WAVES_PER_BLOCK * 32)
void chamfer_wmma(const v2f* __restrict__ ws, float* __restrict__ out) {
    const int lane = threadIdx.x & 31;
    const int wave = blockIdx.x * WAVES_PER_BLOCK + (threadIdx.x >> 5);
    const int tg   = wave % ROW_GROUPS;
    const int dir  = (wave / ROW_GROUPS) & 1;           // 0: q=gt,t=gen ; 1: swapped
    const int b    = wave / (ROW_GROUPS * 2);

    const v2f* cb   = ws + (size_t)b * 4 * NPTS;
    const v2f* q_xy = cb + (dir ? 2 * NPTS : 0);
    const v2f* q_zs = q_xy + NPTS;
    const v2f* t_xy = cb + (dir ? 0 : 2 * NPTS);
    const v2f* t_zs = t_xy + NPTS;

    const bool lo  = lane < 16;
    const int  l16 = lane & 15;

    // Per-lane base pointers selected once by lane half
    const v2f* qp = lo ? q_xy : q_zs;
    const v2f* tp = (lo ? t_xy : t_zs) + l16;

    // A fragments: 4 row tiles of 16 (apply -2 on the query side; K3 = 1)
    v2f Afrag[ROW_TILES_PER_WAVE];
#pragma unroll
    for (int rt = 0; rt < ROW_TILES_PER_WAVE; ++rt) {
        int row = (tg * ROW_TILES_PER_WAVE + rt) * 16 + l16;
        v2f a = qp[row];                    // (x,y) | (z,sq)
        Afrag[rt].x = -2.0f * a.x;          // -2x | -2z
        Afrag[rt].y = lo ? (-2.0f * a.y) : 1.0f;
    }

    v8f czero = {};
    v8f rmin[ROW_TILES_PER_WAVE];
#pragma unroll
    for (int rt = 0; rt < ROW_TILES_PER_WAVE; ++rt)
#pragma unroll
        for (int i = 0; i < 8; ++i) rmin[rt][i] = 3.402823466e+38f;

    auto body = [&](v2f Bfrag) {
#pragma unroll
        for (int rt = 0; rt < ROW_TILES_PER_WAVE; ++rt) {
            v8f d = __builtin_amdgcn_wmma_f32_16x16x4_f32(
                false, Afrag[rt], false, Bfrag, (short)0, czero, false, false);
#pragma unroll
            for (int i = 0; i < 8; ++i) rmin[rt][i] = fmin_asm(rmin[rt][i], d[i]);
        }
    };

    // Software pipeline: issue next tile's b64 load before this tile's WMMAs
    v2f Bcur = tp[0];
    for (int nt = 0; nt < NT - 1; ++nt) {
        v2f Bnext = tp[(nt + 1) * 16];      // prefetch (waited next iteration)
        body(Bcur);
        Bcur = Bnext;
    }
    body(Bcur);                             // peeled last tile

    // min across the 16 lanes of each half
#pragma unroll
    for (int rt = 0; rt < ROW_TILES_PER_WAVE; ++rt)
#pragma unroll
        for (int i = 0; i < 8; ++i) {
            float v = rmin[rt][i];
            v = fmin_asm(v, __shfl_xor(v, 1, 32));
            v = fmin_asm(v, __shfl_xor(v, 2, 32));
            v = fmin_asm(v, __shfl_xor(v, 4, 32));
            v = fmin_asm(v, __shfl_xor(v, 8, 32));
            rmin[rt][i] = v;
        }

    if (l16 == 0) {                 // lane 0 -> rows r, lane 16 -> rows r+8
        const float* qs = (const float*)q_zs;   // sq at odd float offsets
        float s = 0.0f;
#pragma unroll
        for (int rt = 0; rt < ROW_TILES_PER_WAVE; ++rt) {
            int rbase = (tg * ROW_TILES_PER_WAVE + rt) * 16 + (lo ? 0 : 8);
#pragma unroll
            for (int i = 0; i < 8; ++i)
                s += rmin[rt][i] + qs[(rbase + i) * 2 + 1];
        }
        atomicAdd(out, s * (1.0f / BATCH));     // mean over batch of both sums
    }
}

extern "C" void kernel_launch(void* const* d_in, const int* in_sizes, int n_in,
                              void* d_out, int out_size, void* d_ws, size_t ws_size,
                              hipStream_t stream) {
    const float* pred   = (const float*)d_in[0];  // registration_pred [B,C,N]
    const float* gt     = (const float*)d_in[1];  // registration_gt   [B,C,N]
    const float* coords = (const float*)d_in[2];  // coords            [B,C,N]
    float* out = (float*)d_out;
    v2f*   ws  = (v2f*)d_ws;                      // needs 1 MB

    hipMemsetAsync(out, 0, sizeof(float), stream);

    int total = BATCH * NPTS;
    chamfer_pre<<<(total + 255) / 256, 256, 0, stream>>>(pred, gt, coords, ws);

    int waves  = BATCH * 2 * ROW_GROUPS;          // 1024 waves
    int blocks = waves / WAVES_PER_BLOCK;         // 128 blocks x 256 threads
    chamfer_wmma<<<blocks, WAVES_PER_BLOCK * 32, 0, stream>>>(ws, out);
}